// GAT_18846316495041
// MI455X (gfx1250) — compile-verified
//
#include <hip/hip_runtime.h>

typedef __attribute__((ext_vector_type(2))) float v2f;
typedef __attribute__((ext_vector_type(8))) float v8f;

#define N_NODES 8192
#define NCHUNK 16                                // column chunks per row-tile
#define ROW_TILES (N_NODES / 16)                 // 512
#define COLS_PER_CHUNK (N_NODES / NCHUNK)        // 512
#define TILES_PER_CHUNK (COLS_PER_CHUNK / 16)    // 32
#define WAVES_PER_BLOCK 8

// log2(e) / sqrt(8192): exp(d/sqrt(N)) == exp2(d * C_EXP2)
#define C_EXP2 0.0159396789917f

// One wave computes a 16-row x COLS_PER_CHUNK strip of the fused
// att = rownorm(exp(h_i h^T / sqrt(N)) * adj);  h_new = att @ h
// Scores for each 16x16 tile come from one V_WMMA_F32_16X16X4_F32 (K=3 padded to 4).
__global__ __launch_bounds__(256) void gat_head_kernel(
    const float* __restrict__ hsrc,   // [N,3] current features
    const float* __restrict__ adj,    // [N,N] binary mask
    const float* __restrict__ W,      // [3,3] this head's weight (row-major [g][f])
    float* __restrict__ partial)      // [NCHUNK, N, 4] = {rowsum, acc0, acc1, acc2}
{
    const int lane = threadIdx.x & 31;
    const int wave = threadIdx.x >> 5;
    const int gw = blockIdx.x * WAVES_PER_BLOCK + wave;
    const int row_tile = gw >> 4;          // / NCHUNK
    const int chunk    = gw & (NCHUNK - 1);
    const int rbase = row_tile * 16;
    const int cbase0 = chunk * COLS_PER_CHUNK;

    // ---- A fragment: h_i = h @ W^T for my 16 rows (16x4 f32, K=3 padded) ----
    // Layout: lanes 0-15 hold M=lane {K0,K1}, lanes 16-31 hold M=lane-16 {K2,K3}
    const int myrow = rbase + (lane & 15);
    const float x0 = hsrc[myrow * 3 + 0];
    const float x1 = hsrc[myrow * 3 + 1];
    const float x2 = hsrc[myrow * 3 + 2];
    const float hi0 = x0 * W[0] + x1 * W[1] + x2 * W[2];
    const float hi1 = x0 * W[3] + x1 * W[4] + x2 * W[5];
    const float hi2 = x0 * W[6] + x1 * W[7] + x2 * W[8];
    v2f afrag;
    afrag.x = (lane < 16) ? hi0 : hi2;
    afrag.y = (lane < 16) ? hi1 : 0.0f;   // K=3 zero pad

    float rowsum[8];
    float hacc[8][3];
#pragma unroll
    for (int r = 0; r < 8; ++r) {
        rowsum[r] = 0.0f;
        hacc[r][0] = 0.0f; hacc[r][1] = 0.0f; hacc[r][2] = 0.0f;
    }

    // adj element this lane owns inside a tile: row (rbase + r + 8*(lane>>4)), col cb + (lane&15)
    const size_t row_off = (size_t)(rbase + ((lane >> 4) << 3)) * N_NODES + (size_t)(lane & 15);
    const float* __restrict__ aptr = adj + row_off;

#pragma unroll 2
    for (int t = 0; t < TILES_PER_CHUNK; ++t) {
        const int cb = cbase0 + t * 16;
        const int m = cb + (lane & 15);
        const float hm0 = hsrc[m * 3 + 0];
        const float hm1 = hsrc[m * 3 + 1];
        const float hm2 = hsrc[m * 3 + 2];
        // B fragment (4x16 f32): lanes 0-15 = rows K0,K1 of col N=lane; lanes 16-31 = K2,K3
        v2f bfrag;
        bfrag.x = (lane < 16) ? hm0 : hm2;
        bfrag.y = (lane < 16) ? hm1 : 0.0f;

        const float* __restrict__ a2 = aptr + cb;
        float av[8];
#pragma unroll
        for (int r = 0; r < 8; ++r)
            av[r] = a2[(size_t)r * N_NODES];

        v8f c = {};
        // D = A(16x4) x B(4x16): 16x16 raw score tile
        v8f d = __builtin_amdgcn_wmma_f32_16x16x4_f32(
            false, afrag, false, bfrag, (short)0, c, false, false);

#pragma unroll
        for (int r = 0; r < 8; ++r) {
            // adj is {0,1}: multiply implements the mask exactly
            const float s = av[r] * __builtin_amdgcn_exp2f(d[r] * C_EXP2);
            rowsum[r]  += s;
            hacc[r][0] += s * hm0;
            hacc[r][1] += s * hm1;
            hacc[r][2] += s * hm2;
        }
    }

    // ---- reduce across the 16 lanes of each half (rows r in lanes 0-15, r+8 in 16-31) ----
#pragma unroll
    for (int r = 0; r < 8; ++r) {
        float s  = rowsum[r];
        float a0 = hacc[r][0], a1 = hacc[r][1], a2 = hacc[r][2];
#pragma unroll
        for (int off = 8; off >= 1; off >>= 1) {
            s  += __shfl_xor(s,  off, 32);
            a0 += __shfl_xor(a0, off, 32);
            a1 += __shfl_xor(a1, off, 32);
            a2 += __shfl_xor(a2, off, 32);
        }
        if (lane == 0 || lane == 16) {
            const int row = rbase + r + ((lane >> 4) << 3);
            float4 o = make_float4(s, a0, a1, a2);
            *(float4*)(partial + ((size_t)chunk * N_NODES + row) * 4) = o;
        }
    }
}

// h_new[n] = (sum_chunk acc) / (sum_chunk rowsum)
__global__ __launch_bounds__(256) void gat_combine_kernel(
    const float* __restrict__ partial, float* __restrict__ hdst)
{
    const int n = blockIdx.x * blockDim.x + threadIdx.x;
    if (n >= N_NODES) return;
    float s = 0.0f, a0 = 0.0f, a1 = 0.0f, a2 = 0.0f;
#pragma unroll
    for (int c = 0; c < NCHUNK; ++c) {
        const float4 p = *(const float4*)(partial + ((size_t)c * N_NODES + n) * 4);
        s += p.x; a0 += p.y; a1 += p.z; a2 += p.w;
    }
    hdst[n * 3 + 0] = a0 / s;
    hdst[n * 3 + 1] = a1 / s;
    hdst[n * 3 + 2] = a2 / s;
}

// logits[n,c] = concat(x, h1, h2, h3)[n,:] . W_out[c,:]
__global__ __launch_bounds__(256) void gat_out_kernel(
    const float* __restrict__ x,
    const float* __restrict__ h0, const float* __restrict__ h1,
    const float* __restrict__ h2,
    const float* __restrict__ Wout,   // [8,12]
    float* __restrict__ out)          // [N,8]
{
    const int n = blockIdx.x * blockDim.x + threadIdx.x;
    if (n >= N_NODES) return;
    float f[12];
#pragma unroll
    for (int k = 0; k < 3; ++k) {
        f[k]     = x[n * 3 + k];
        f[3 + k] = h0[n * 3 + k];
        f[6 + k] = h1[n * 3 + k];
        f[9 + k] = h2[n * 3 + k];
    }
#pragma unroll
    for (int c = 0; c < 8; ++c) {
        float acc = 0.0f;
#pragma unroll
        for (int k = 0; k < 12; ++k) acc += f[k] * Wout[c * 12 + k];
        out[n * 8 + c] = acc;
    }
}

extern "C" void kernel_launch(void* const* d_in, const int* in_sizes, int n_in,
                              void* d_out, int out_size, void* d_ws, size_t ws_size,
                              hipStream_t stream) {
    const float* x       = (const float*)d_in[0];   // [1,N,3]
    const float* adj     = (const float*)d_in[1];   // [1,N,N]
    const float* W_heads = (const float*)d_in[2];   // [4,3,3]
    const float* W_out   = (const float*)d_in[3];   // [8,12]
    float* out = (float*)d_out;

    float* ws = (float*)d_ws;
    float* hbuf[3] = { ws, ws + N_NODES * 3, ws + 2 * N_NODES * 3 };
    float* partial = ws + 3 * N_NODES * 3;          // NCHUNK*N*4 floats (~2 MB)

    const int head_blocks = (ROW_TILES * NCHUNK) / WAVES_PER_BLOCK;  // 1024
    const int pt_blocks = N_NODES / 256;

    const float* hsrc = x;
    for (int i = 0; i < 3; ++i) {                   // heads[:-1]
        gat_head_kernel<<<head_blocks, 256, 0, stream>>>(hsrc, adj, W_heads + i * 9, partial);
        gat_combine_kernel<<<pt_blocks, 256, 0, stream>>>(partial, hbuf[i]);
        hsrc = hbuf[i];
    }
    gat_out_kernel<<<pt_blocks, 256, 0, stream>>>(x, hbuf[0], hbuf[1], hbuf[2], W_out, out);
}